// Point2DirSum_37701222924357
// MI455X (gfx1250) — compile-verified
//
#include <hip/hip_runtime.h>

// Problem constants (from reference): input (128,16,128,128) f32,
// output (128,64,128,128) f32 = [hor, ver, diag, adiag] along channel axis.
#define NB   128
#define C    16
#define H    128
#define W    128
#define PAD  2          // WIN_LENGTH/2
#define TH   32         // output rows per block tile
#define TROWS (TH + 2*PAD)   // 36 rows incl. halo
#define LS   136        // LDS row stride (floats): keeps interior 16B-aligned
#define COL0 4          // LDS column holding global column 0

#if defined(__HIP_DEVICE_COMPILE__) && __has_builtin(__builtin_amdgcn_global_load_async_to_lds_b128)
#define USE_ASYNC_LDS 1
#endif

// Exact pointee type expected by the async-to-LDS builtins (per hipcc diagnostic):
typedef int v4i_t __attribute__((vector_size(16)));
typedef __attribute__((address_space(1))) v4i_t* g_v4i_ptr;   // global
typedef __attribute__((address_space(3))) v4i_t* l_v4i_ptr;   // LDS

__global__ __launch_bounds__(256)
void dirsum5_kernel(const float* __restrict__ in, float* __restrict__ out) {
    __shared__ __align__(16) float tile[TROWS * LS];

    const int bid = blockIdx.x;          // 0 .. NB*C*(H/TH)-1
    const int t   = bid & 3;             // row-tile index (H/TH == 4)
    const int c   = (bid >> 2) & (C - 1);
    const int n   = bid >> 6;
    const int h0  = t * TH;
    const int tid = threadIdx.x;

    const float* gplane = in + (size_t)(n * C + c) * (H * W);

    // ---- 1) zero halo / out-of-bounds cells (disjoint from async targets) ----
    for (int i = tid; i < TROWS * LS; i += 256) {
        const int r    = i / LS;
        const int cc   = i - r * LS;
        const int grow = h0 + r - PAD;
        const bool interior = (grow >= 0) && (grow < H) && (cc >= COL0) && (cc < COL0 + W);
        if (!interior) tile[i] = 0.0f;
    }

    // ---- 2) interior rows: async copy global -> LDS, 16B per op ----
    for (int j = tid; j < TROWS * (W / 4); j += 256) {
        const int r    = j >> 5;          // W/4 == 32 quads per row
        const int q    = j & 31;
        const int grow = h0 + r - PAD;
        if (grow >= 0 && grow < H) {
            const float* gp = gplane + (size_t)grow * W + q * 4;
            float*       lp = &tile[r * LS + COL0 + q * 4];
#ifdef USE_ASYNC_LDS
            __builtin_amdgcn_global_load_async_to_lds_b128(
                (g_v4i_ptr)(v4i_t*)gp,
                (l_v4i_ptr)(v4i_t*)lp,
                /*offset=*/0, /*cpol=*/0);
#else
            *(float4*)lp = *(const float4*)gp;
#endif
        }
    }
#ifdef USE_ASYNC_LDS
#if __has_builtin(__builtin_amdgcn_s_wait_asynccnt)
    __builtin_amdgcn_s_wait_asynccnt(0);
#else
    asm volatile("s_wait_asynccnt 0" ::: "memory");
#endif
#endif
    __syncthreads();

    // ---- 3) compute all 4 directions from LDS, float4 outputs ----
    const size_t planeHW = (size_t)H * W;
    float* o0 = out + ((size_t)n * 4 * C + 0 * C + c) * planeHW;  // horizontal
    float* o1 = o0 + (size_t)C * planeHW;                         // vertical
    float* o2 = o1 + (size_t)C * planeHW;                         // diagonal
    float* o3 = o2 + (size_t)C * planeHW;                         // anti-diagonal

    for (int k = tid; k < TH * (W / 4); k += 256) {
        const int hh = k >> 5;
        const int w  = (k & 31) * 4;
        const int r  = hh + PAD;          // LDS center row

        float ah[4] = {0.f, 0.f, 0.f, 0.f};
        float av[4] = {0.f, 0.f, 0.f, 0.f};
        float ad[4] = {0.f, 0.f, 0.f, 0.f};
        float aa[4] = {0.f, 0.f, 0.f, 0.f};

        // 5x8 patch: rows r-2..r+2, lds cols (COL0+w-2) .. +7 covers all dirs
#pragma unroll
        for (int dr = 0; dr < 5; ++dr) {
            const int rb = (r - 2 + dr) * LS + (COL0 + w - 2);
            float p[8];
#pragma unroll
            for (int j = 0; j < 8; ++j) p[j] = tile[rb + j];
#pragma unroll
            for (int j0 = 0; j0 < 4; ++j0) {
                av[j0] += p[j0 + 2];
                ad[j0] += p[j0 + dr];
                aa[j0] += p[j0 + 4 - dr];
            }
            if (dr == 2) {
#pragma unroll
                for (int j0 = 0; j0 < 4; ++j0) {
#pragma unroll
                    for (int dw = 0; dw < 5; ++dw) ah[j0] += p[j0 + dw];
                }
            }
        }

        const size_t off = (size_t)(h0 + hh) * W + w;
        *(float4*)(o0 + off) = make_float4(ah[0], ah[1], ah[2], ah[3]);
        *(float4*)(o1 + off) = make_float4(av[0], av[1], av[2], av[3]);
        *(float4*)(o2 + off) = make_float4(ad[0], ad[1], ad[2], ad[3]);
        *(float4*)(o3 + off) = make_float4(aa[0], aa[1], aa[2], aa[3]);
    }
}

extern "C" void kernel_launch(void* const* d_in, const int* in_sizes, int n_in,
                              void* d_out, int out_size, void* d_ws, size_t ws_size,
                              hipStream_t stream) {
    const float* in  = (const float*)d_in[0];
    float*       out = (float*)d_out;
    const int nblocks = NB * C * (H / TH);   // 8192
    dirsum5_kernel<<<dim3(nblocks), dim3(256), 0, stream>>>(in, out);
}